// Generator3DLUT_zero_20744692039901
// MI455X (gfx1250) — compile-verified
//
#include <hip/hip_runtime.h>

// Trilinear 3D-LUT apply, tuned for MI455X (gfx1250):
//  - LUT packed as f16x4 cells in LDS (287.5 KB -- only possible with CDNA5's 320KB LDS)
//  - LUT staged into LDS via Tensor Data Mover (tensor_load_to_lds + s_wait_tensorcnt)
//  - x streamed as float4 (b128), one ds_load_b64 per LUT corner (3 channels at once)
// Bandwidth-bound: ~201 MB total HBM traffic -> ~8.6us at 23.3 TB/s.

#define DIM    33
#define DIM2   (DIM * DIM)            // 1089
#define NCELL  (DIM * DIM * DIM)      // 35937
#define CHUNK  2048                   // staging chunk (floats per channel)
#define TABLE_BYTES (NCELL * 8)       // 287496 B: f16x4 per cell
#define STAGE_BYTES (3 * CHUNK * 4)   // 24576 B: fp32 staging for 3 channels
#define LDS_BYTES   (TABLE_BYTES + STAGE_BYTES)  // 312072 B <= 320KB

typedef __attribute__((ext_vector_type(4))) _Float16    h4;
typedef __attribute__((ext_vector_type(4))) unsigned    u32x4;
typedef __attribute__((ext_vector_type(8))) int         i32x8;
typedef __attribute__((ext_vector_type(4))) int         i32x4;

#ifndef HAVE_TDM
#if defined(__has_builtin)
#if __has_builtin(__builtin_amdgcn_tensor_load_to_lds)
#define HAVE_TDM 1
#else
#define HAVE_TDM 0
#endif
#else
#define HAVE_TDM 0
#endif
#endif

extern __shared__ char smem[];

#if HAVE_TDM
// 1-D TDM copy: nelem fp32 from global -> LDS at byte offset lds_off.
// Descriptor per CDNA5 ISA (D# groups 0/1; groups 2/3 zero).
// This toolchain (clang-23 / therock-10) uses the 6-arg builtin:
//   (u32x4 g0, i32x8 g1, i32x4 g2, i32x4 g3, i32x8 extra, i32 cpol)
__device__ __forceinline__ void tdm_load_1d(unsigned lds_off, const float* gsrc, int nelem) {
  unsigned long long ga = (unsigned long long)(size_t)gsrc;
  unsigned un = (unsigned)nelem;
  u32x4 g0 = {
    1u,                                                    // [1:0] count=1 (valid), rest 0
    lds_off,                                               // [63:32] LDS byte address
    (unsigned)(ga & 0xFFFFFFFFull),                        // [95:64] global addr lo
    (unsigned)((ga >> 32) & 0x01FFFFFFull) | (2u << 30)    // [120:96] addr hi, [127:126] type=2
  };
  i32x8 g1 = {
    (int)(2u << 16),                                       // workgroup_mask=0, data_size=2 (4B)
    (int)((un & 0xFFFFu) << 16),                           // tensor_dim0[15:0] @bits[63:48]
    (int)(((un >> 16) & 0xFFFFu) | (1u << 16)),            // tensor_dim0[31:16]; tensor_dim1=1
    (int)((un & 0xFFFFu) << 16),                           // tile_dim0 = nelem @bits[127:112]
    0,                                                     // tile_dim1=0, tile_dim2=0 (1-D tile)
    (int)un,                                               // tensor_dim0_stride[31:0]
    0, 0                                                   // stride hi / dim1 stride = 0
  };
  i32x4 gz4 = {0, 0, 0, 0};
  i32x8 gz8 = {0, 0, 0, 0, 0, 0, 0, 0};
  __builtin_amdgcn_tensor_load_to_lds(g0, g1, gz4, gz4, gz8, 0);
}
#endif

struct f3 { float x, y, z; };

__device__ __forceinline__ f3 h4tof3(h4 v) {
  f3 r; r.x = (float)v.x; r.y = (float)v.y; r.z = (float)v.z; return r;
}
__device__ __forceinline__ f3 lerp3(f3 a, f3 b, float t) {
  f3 r;
  r.x = fmaf(t, b.x - a.x, a.x);
  r.y = fmaf(t, b.y - a.y, a.y);
  r.z = fmaf(t, b.z - a.z, a.z);
  return r;
}

// One pixel: 8 ds_load_b64 gathers + 7 trilerp steps, all 3 channels together.
__device__ __forceinline__ void sample1(const h4* __restrict__ tab,
                                        float r, float g, float b,
                                        float& oR, float& oG, float& oB) {
  const float S = (float)(DIM - 1);
  float rf = r * S, gf = g * S, bf = b * S;
  int ri = (int)floorf(rf); ri = ri < 0 ? 0 : (ri > DIM - 2 ? DIM - 2 : ri);
  int gi = (int)floorf(gf); gi = gi < 0 ? 0 : (gi > DIM - 2 ? DIM - 2 : gi);
  int bi = (int)floorf(bf); bi = bi < 0 ? 0 : (bi > DIM - 2 ? DIM - 2 : bi);
  float rd = rf - (float)ri, gd = gf - (float)gi, bd = bf - (float)bi;
  int idx = ri + gi * DIM + bi * DIM2;              // matches LUT flat r + g*dim + b*dim^2
  f3 c000 = h4tof3(tab[idx]);
  f3 c100 = h4tof3(tab[idx + 1]);
  f3 c010 = h4tof3(tab[idx + DIM]);
  f3 c110 = h4tof3(tab[idx + DIM + 1]);
  f3 c001 = h4tof3(tab[idx + DIM2]);
  f3 c101 = h4tof3(tab[idx + DIM2 + 1]);
  f3 c011 = h4tof3(tab[idx + DIM2 + DIM]);
  f3 c111 = h4tof3(tab[idx + DIM2 + DIM + 1]);
  f3 c00 = lerp3(c000, c100, rd), c10 = lerp3(c010, c110, rd);
  f3 c01 = lerp3(c001, c101, rd), c11 = lerp3(c011, c111, rd);
  f3 c0 = lerp3(c00, c10, gd),    c1 = lerp3(c01, c11, gd);
  f3 c  = lerp3(c0, c1, bd);
  oR = c.x; oG = c.y; oB = c.z;
}

__global__ __launch_bounds__(1024)
void trilinear_lut3d_kernel(const float* __restrict__ LUT,
                            const float* __restrict__ x,
                            float* __restrict__ out,
                            int npix, int HW) {
  h4*    tab   = (h4*)smem;
  float* stage = (float*)(smem + TABLE_BYTES);
  const int tid  = threadIdx.x;
  const int nthr = blockDim.x;

  // ---- Stage LUT (3 x 35937 fp32) into LDS as packed f16x4 cells ----
  for (int c0 = 0; c0 < NCELL; c0 += CHUNK) {
    const int n = (NCELL - c0) < CHUNK ? (NCELL - c0) : CHUNK;
#if HAVE_TDM
    if (tid < 32) {  // wave 0 issues the async tensor DMAs
      #pragma unroll
      for (int ch = 0; ch < 3; ++ch) {
        unsigned lds_off = (unsigned)(size_t)(void*)(stage + ch * CHUNK);
        tdm_load_1d(lds_off, LUT + ch * NCELL + c0, n);
      }
      __builtin_amdgcn_s_wait_tensorcnt(0);
    }
    __syncthreads();
    for (int t = tid; t < n; t += nthr) {
      h4 v;
      v.x = (_Float16)stage[t];
      v.y = (_Float16)stage[CHUNK + t];
      v.z = (_Float16)stage[2 * CHUNK + t];
      v.w = (_Float16)0.f;
      tab[c0 + t] = v;
    }
#else
    for (int t = tid; t < n; t += nthr) {
      h4 v;
      v.x = (_Float16)LUT[c0 + t];
      v.y = (_Float16)LUT[NCELL + c0 + t];
      v.z = (_Float16)LUT[2 * NCELL + c0 + t];
      v.w = (_Float16)0.f;
      tab[c0 + t] = v;
    }
#endif
    __syncthreads();
  }

  // ---- Stream pixels: 4 per thread via float4 (b128) loads/stores ----
  const int nvec   = npix >> 2;
  const int stride = (int)(gridDim.x * nthr);
  for (int i = (int)(blockIdx.x * nthr) + tid; i < nvec; i += stride) {
    const int  q    = i << 2;
    const int  nb   = q / HW;                    // batch index
    const long long base = (long long)q + 2LL * nb * HW;  // = (nb*3)*HW + p
    __builtin_prefetch(x + base + (long long)stride * 4, 0, 1);  // global_prefetch_b8
    const float4 r4 = *(const float4*)(x + base);
    const float4 g4 = *(const float4*)(x + base + HW);
    const float4 b4 = *(const float4*)(x + base + 2 * HW);
    float4 oR, oG, oB;
    sample1(tab, r4.x, g4.x, b4.x, oR.x, oG.x, oB.x);
    sample1(tab, r4.y, g4.y, b4.y, oR.y, oG.y, oB.y);
    sample1(tab, r4.z, g4.z, b4.z, oR.z, oG.z, oB.z);
    sample1(tab, r4.w, g4.w, b4.w, oR.w, oG.w, oB.w);
    *(float4*)(out + base)           = oR;
    *(float4*)(out + base + HW)      = oG;
    *(float4*)(out + base + 2 * HW)  = oB;
  }
}

extern "C" void kernel_launch(void* const* d_in, const int* in_sizes, int n_in,
                              void* d_out, int out_size, void* d_ws, size_t ws_size,
                              hipStream_t stream) {
  const float* LUT = (const float*)d_in[0];  // (3, 33, 33, 33) fp32
  const float* x   = (const float*)d_in[1];  // (B, 3, 1024, 1024) fp32
  float* out = (float*)d_out;

  const int HW   = 1024 * 1024;
  const int npix = in_sizes[1] / 3;          // B * H * W
  const int nvec = npix >> 2;

  const int threads = 1024;                  // 32 waves -> fills a WGP (1 WG/WGP due to LDS)
  int blocks = 512;                          // grid-stride; amortizes LUT staging per block
  int maxb = (nvec + threads - 1) / threads;
  if (blocks > maxb) blocks = maxb;
  if (blocks < 1) blocks = 1;

  (void)hipFuncSetAttribute(reinterpret_cast<const void*>(&trilinear_lut3d_kernel),
                            hipFuncAttributeMaxDynamicSharedMemorySize, LDS_BYTES);
  trilinear_lut3d_kernel<<<blocks, threads, LDS_BYTES, stream>>>(LUT, x, out, npix, HW);
}